// Emb_model_59854664237698
// MI455X (gfx1250) — compile-verified
//
#include <hip/hip_runtime.h>

// GraphSAGE 2-layer forward for MI455X (gfx1250), fp32 throughout.
// N=50000 nodes, E=800000 edges, D=64 features/hidden.

typedef __attribute__((ext_vector_type(2))) float v2f;
typedef __attribute__((ext_vector_type(8))) float v8f;

#define GN 50000
#define GE 800000
#define GD 64
// LDS row stride (in floats) for one K-pair row of a weight matrix.
// 128 data words + 32 pad -> stride % 64 == 32, so the two half-wave K-pair
// rows map to disjoint bank sets (0-31 vs 32-63): conflict-free ds_load_b64.
#define WROW 160

// ---------------------------------------------------------------------------
// Degree count: cnt[dst[e]] += 1.0  (f32 atomics, L2-resident target)
// ---------------------------------------------------------------------------
__global__ __launch_bounds__(256) void count_edges_k(const int* __restrict__ dst,
                                                     float* __restrict__ cnt,
                                                     int e_total) {
  int e = blockIdx.x * 256 + threadIdx.x;
  if (e < e_total) atomicAdd(&cnt[dst[e]], 1.0f);
}

// ---------------------------------------------------------------------------
// Scatter-add: agg[dst[e]][f] += feat[src[e]][f]
// One thread per (edge, feature); wave32 covers 32 consecutive features ->
// 128B coalesced gather loads; atomic adds land in L2 (agg = 12.8MB << 192MB).
// ---------------------------------------------------------------------------
__global__ __launch_bounds__(256) void scatter_add_k(const int* __restrict__ src,
                                                     const int* __restrict__ dst,
                                                     const float* __restrict__ feat,
                                                     float* __restrict__ agg,
                                                     int e_total) {
  long long t = (long long)blockIdx.x * 256 + threadIdx.x;
  int e = (int)(t >> 6);
  int f = (int)(t & 63);
  if (e < e_total) {
    int s = src[e];
    int d = dst[e];
    atomicAdd(&agg[(long long)d * GD + f], feat[(long long)s * GD + f]);
  }
}

// ---------------------------------------------------------------------------
// Fused SAGEConv GEMM: h[row] = relu( (agg[row]/max(cnt,1)) @ Wl^T + bl
//                                      + xr[row] @ Wr^T )
// One wave per 16-row tile; full fp32 via V_WMMA_F32_16X16X4_F32.
// Weights staged in LDS as interleaved K-pairs: WS[k/2][n] = {W[n][k],W[n][k+1]}
// so each B fragment is ONE aligned ds_load_b64 straight into the WMMA operand
// pair (no register re-pairing shuffles), bank-conflict-free via WROW padding.
// K=64 -> 16 k-steps x 4 N-tiles x 2 matrices = 128 WMMA ops per wave.
// ---------------------------------------------------------------------------
__global__ __launch_bounds__(256) void sage_gemm_relu_k(
    const float* __restrict__ agg, const float* __restrict__ cnt,
    const float* __restrict__ xr,
    const float* __restrict__ Wl, const float* __restrict__ Wr,
    const float* __restrict__ bl,
    float* __restrict__ hout, int ntiles)
{
  __shared__ float WlS[(GD / 2) * WROW];
  __shared__ float WrS[(GD / 2) * WROW];
  for (int i = threadIdx.x; i < GD * GD; i += 256) {
    int j = i >> 6;   // output-channel row of W
    int k = i & 63;   // input-channel col of W
    int o = (k >> 1) * WROW + j * 2 + (k & 1);
    WlS[o] = Wl[i];
    WrS[o] = Wr[i];
  }
  __syncthreads();

  int wave = threadIdx.x >> 5;
  int tile = blockIdx.x * 8 + wave;          // wave-uniform
  if (tile < ntiles) {
    int lane = threadIdx.x & 31;
    int half = lane >> 4;                    // 0: lanes 0-15, 1: lanes 16-31
    int l16  = lane & 15;
    int row  = tile * 16 + l16;              // A-matrix row owned by this lane

    float inv = 1.0f / fmaxf(cnt[row], 1.0f);  // exact div, once per wave
    const float* aggRow = agg + (long long)row * GD;
    const float* xRow   = xr  + (long long)row * GD;

    v8f acc[4] = {};                         // 4 N-tiles of 16x16 f32

    for (int k0 = 0; k0 < GD; k0 += 4) {
      // A fragment (32-bit 16x4): lanes 0-15 hold K=k0,k0+1; lanes 16-31 hold
      // K=k0+2,k0+3 -> one contiguous float2 per lane.
      int ka = k0 + half * 2;
      v2f am = *(const v2f*)(aggRow + ka);
      am *= inv;                             // mean aggregation on the fly
      v2f ar = *(const v2f*)(xRow + ka);
      int kb = (ka >> 1) * WROW;             // K-pair row base in LDS
#pragma unroll
      for (int t = 0; t < 4; ++t) {
        int n = t * 16 + l16;                // B-matrix column owned by lane
        v2f bL = *(const v2f*)&WlS[kb + n * 2];   // single ds_load_b64
        v2f bR = *(const v2f*)&WrS[kb + n * 2];
        acc[t] = __builtin_amdgcn_wmma_f32_16x16x4_f32(
            false, am, false, bL, (short)0, acc[t], false, false);
        acc[t] = __builtin_amdgcn_wmma_f32_16x16x4_f32(
            false, ar, false, bR, (short)0, acc[t], false, false);
      }
    }

    // Epilogue: bias + ReLU + store.
    // C/D layout: VGPR v holds row (half*8 + v), col l16 -> per (t,v) the
    // 32 lanes store two contiguous 64B segments.
#pragma unroll
    for (int t = 0; t < 4; ++t) {
      int n = t * 16 + l16;
      float bias = bl[n];
#pragma unroll
      for (int v = 0; v < 8; ++v) {
        int mr = half * 8 + v;
        float val = fmaxf(acc[t][v] + bias, 0.0f);
        hout[(long long)(tile * 16 + mr) * GD + n] = val;
      }
    }
  }
}

// ---------------------------------------------------------------------------
// Final head: out[i] = dot(h2[i], Wf) + bf   (float4 vectorized)
// ---------------------------------------------------------------------------
__global__ __launch_bounds__(256) void final_head_k(const float* __restrict__ h,
                                                    const float* __restrict__ Wf,
                                                    const float* __restrict__ bf,
                                                    float* __restrict__ out,
                                                    int n_total) {
  int i = blockIdx.x * 256 + threadIdx.x;
  if (i >= n_total) return;
  const float4* r = (const float4*)(h + (long long)i * GD);
  const float4* w = (const float4*)Wf;
  float s = 0.0f;
#pragma unroll
  for (int q = 0; q < GD / 4; ++q) {
    float4 a = r[q], b = w[q];
    s += a.x * b.x + a.y * b.y + a.z * b.z + a.w * b.w;
  }
  out[i] = s + bf[0];
}

// ---------------------------------------------------------------------------
// Host-side launch. Inputs (dict order):
//   0:x [N,64] f32   1:edge_index [2,E] i32   2:Wl1 [64,64]  3:bl1 [64]
//   4:Wr1 [64,64]    5:Wl2 [64,64]  6:bl2 [64]  7:Wr2 [64,64]
//   8:Wf [1,64]      9:bf [1]
// Output: [N,1] f32.
// Workspace: agg (N*64 f32) | cnt (N f32) | h (N*64 f32)  = ~25.8 MB
// ---------------------------------------------------------------------------
extern "C" void kernel_launch(void* const* d_in, const int* in_sizes, int n_in,
                              void* d_out, int out_size, void* d_ws, size_t ws_size,
                              hipStream_t stream) {
  (void)in_sizes; (void)n_in; (void)out_size; (void)ws_size;

  const float* x   = (const float*)d_in[0];
  const int*   ei  = (const int*)d_in[1];
  const int*   src = ei;        // edge_index[0]
  const int*   dst = ei + GE;   // edge_index[1]
  const float* Wl1 = (const float*)d_in[2];
  const float* bl1 = (const float*)d_in[3];
  const float* Wr1 = (const float*)d_in[4];
  const float* Wl2 = (const float*)d_in[5];
  const float* bl2 = (const float*)d_in[6];
  const float* Wr2 = (const float*)d_in[7];
  const float* Wf  = (const float*)d_in[8];
  const float* bf  = (const float*)d_in[9];

  char*  ws  = (char*)d_ws;
  float* agg = (float*)ws;                                          // N*64 f32
  float* cnt = (float*)(ws + (size_t)GN * GD * 4);                  // N f32
  float* h   = (float*)(ws + (size_t)GN * GD * 4 + (size_t)GN * 4); // N*64 f32

  const int ntiles = GN / 16;                 // 3125, exact
  const int gemm_blocks = (ntiles + 7) / 8;   // 8 waves (tiles) per block
  const int scat_blocks = (GE * GD) / 256;    // one thread per (edge,feat)

  // Layer 1
  hipMemsetAsync(agg, 0, (size_t)GN * GD * 4, stream);
  hipMemsetAsync(cnt, 0, (size_t)GN * 4, stream);
  count_edges_k<<<(GE + 255) / 256, 256, 0, stream>>>(dst, cnt, GE);
  scatter_add_k<<<scat_blocks, 256, 0, stream>>>(src, dst, x, agg, GE);
  sage_gemm_relu_k<<<gemm_blocks, 256, 0, stream>>>(agg, cnt, x, Wl1, Wr1, bl1, h, ntiles);

  // Layer 2 (in-place h update is safe: each wave reads/writes only its own
  // 16-row tile, and all its reads precede its writes)
  hipMemsetAsync(agg, 0, (size_t)GN * GD * 4, stream);
  scatter_add_k<<<scat_blocks, 256, 0, stream>>>(src, dst, h, agg, GE);
  sage_gemm_relu_k<<<gemm_blocks, 256, 0, stream>>>(agg, cnt, h, Wl2, Wr2, bl2, h, ntiles);

  // Head
  final_head_k<<<(GN + 255) / 256, 256, 0, stream>>>(h, Wf, bf, (float*)d_out, GN);
}